// DecomNet_RTV_68624987456229
// MI455X (gfx1250) — compile-verified
//
#include <hip/hip_runtime.h>
#include <hip/hip_bf16.h>
#include <math.h>

// ---------------------------------------------------------------------------
// Types for WMMA f32 16x16x4 (CDNA5 / gfx1250, wave32)
// A: 16x4  -> 2 VGPRs/lane (v2f). B: 4x16 -> 2 VGPRs/lane. C/D: 16x16 -> v8f.
// ---------------------------------------------------------------------------
typedef float v2f __attribute__((ext_vector_type(2)));
typedef float v4f __attribute__((ext_vector_type(4)));
typedef float v8f __attribute__((ext_vector_type(8)));

typedef unsigned int u32x4 __attribute__((ext_vector_type(4)));
typedef int          i32x4 __attribute__((ext_vector_type(4)));
typedef int          i32x8 __attribute__((ext_vector_type(8)));

#define N512      512
#define PLANE     (512LL * 512LL)     // one image plane, floats
#define NIMG      8LL
#define IMGSET    (NIMG * PLANE)      // 8 planes
#define K1_ITERS  10
#define EPS_C     0.001f
#define BN_EPS_C  1e-5f

#if defined(__HIP_DEVICE_COMPILE__) && __has_builtin(__builtin_amdgcn_tensor_load_to_lds)
#define USE_TDM 1
#else
#define USE_TDM 0
#endif

#define LDA_PAD 18   // A rows padded 16 -> 18 floats (TDM pad: 2 DW per 16 DW)

#if USE_TDM
// ---------------------------------------------------------------------------
// Issue one TDM 2-D tile load: global (row-major, stride 512 floats) -> LDS.
// D# packing per CDNA5 ISA 8.3/8.4. Groups 2/3 zero (<=2D tensor).
// tile_d0 = contiguous dim (floats), tile_d1 = rows.
// pad_en=1 adds 2 DWORDs of LDS padding every 16 DWORDs (16->18 rows for A).
// NOTE: this toolchain uses the 6-arg builtin form:
//   (u32x4 g0, i32x8 g1, i32x4 g2, i32x4 g3, i32x8 extra, i32 cpol)
// ---------------------------------------------------------------------------
__device__ __forceinline__ void tdm_load_tile_2d(unsigned int lds_byte_addr,
                                                 const float* gptr,
                                                 unsigned int tile_d0,
                                                 unsigned int tile_d1,
                                                 int pad_en)
{
    unsigned long long ga = (unsigned long long)(uintptr_t)gptr;

    union { unsigned long long q[2]; u32x4 v; } g0;
    g0.q[0] = 1ull                                            // count = 1
            | ((unsigned long long)lds_byte_addr << 32);      // lds_addr [63:32]
    g0.q[1] = (ga & ((1ull << 57) - 1))                       // global_addr [120:64]
            | (2ull << 62);                                   // type = 2 [127:126]

    union { unsigned long long q[4]; i32x8 v; } g1;
    g1.q[0] = g1.q[1] = g1.q[2] = g1.q[3] = 0;
    unsigned long long flags = (2ull << 16);                  // data_size = 4B
    if (pad_en)
        flags |= (1ull << 20)    // pad_enable
               | (3ull << 22)    // pad_interval: every 16 DWORDs
               | (1ull << 25);   // pad_amount:   2 DWORDs
    const unsigned long long td0 = 512, td1 = 512, st = 512;
    g1.q[0] = flags
            | (td0 << 48);                                    // tensor_dim0 [79:48] lo
    g1.q[1] = (td0 >> 16)                                     // tensor_dim0 hi
            | (td1 << 16)                                     // tensor_dim1 [111:80]
            | ((unsigned long long)tile_d0 << 48);            // tile_dim0 [127:112]
    g1.q[2] = (unsigned long long)tile_d1                     // tile_dim1 [143:128]
            | (st << 32);                                     // dim0_stride [207:160] lo
    g1.q[3] = (st >> 32)                                      // dim0_stride hi
            | (st << 16);                                     // dim1_stride [255:208]

    i32x4 zz4 = {0, 0, 0, 0};
    i32x8 zz8 = {0, 0, 0, 0, 0, 0, 0, 0};
    __builtin_amdgcn_tensor_load_to_lds(g0.v, g1.v, zz4, zz4, zz8, 0);
}
#endif

// ===========================================================================
// GEMM: D[z] (+)= alpha * A[z] * B[z],  all 512x512 row-major f32.
// Block: 128 threads = 4 waves; block tile 64x64; wave tile 32x32 (2x2 WMMA).
// LDS tiles staged by the Tensor Data Mover (wave 0 issues descriptors,
// waits on TENSORcnt, workgroup barrier releases compute waves).
// ===========================================================================
__global__ __launch_bounds__(128) void gemm512_wmma(
    float* __restrict__ D, const float* __restrict__ A, const float* __restrict__ B,
    float alpha, int accum, long long sD, long long sA, long long sB)
{
    __shared__ float ldsA[64 * LDA_PAD];
    __shared__ float ldsB[16 * 64];

    const int z = blockIdx.z;
    D += (long long)z * sD;
    A += (long long)z * sA;
    B += (long long)z * sB;

    const int tileM = blockIdx.y * 64;
    const int tileN = blockIdx.x * 64;

    const int tid  = threadIdx.x;
    const int wv   = tid >> 5;
    const int lane = tid & 31;
    const int half = lane >> 4;    // 0: lanes 0-15, 1: lanes 16-31
    const int lm   = lane & 15;
    const int wm   = (wv >> 1) * 32;   // wave row origin in block tile
    const int wn   = (wv & 1)  * 32;   // wave col origin in block tile

#if USE_TDM
    const unsigned int ldsA_addr = (unsigned int)(uintptr_t)&ldsA[0];
    const unsigned int ldsB_addr = (unsigned int)(uintptr_t)&ldsB[0];
#endif

    v8f acc00 = {}, acc01 = {}, acc10 = {}, acc11 = {};

    for (int k0 = 0; k0 < N512; k0 += 16) {
#if USE_TDM
        // ---- TDM-staged tiles: A (64 rows x 16, padded to 18), B (16 x 64) ----
        if (wv == 0) {
            tdm_load_tile_2d(ldsA_addr, A + (long long)tileM * N512 + k0, 16u, 64u, 1);
            tdm_load_tile_2d(ldsB_addr, B + (long long)k0 * N512 + tileN, 64u, 16u, 0);
            __builtin_amdgcn_s_wait_tensorcnt(0);
        }
        __syncthreads();
#else
        // ---- manual staging fallback ----
        #pragma unroll
        for (int rep = 0; rep < 2; ++rep) {
            int idx4 = tid + rep * 128;          // 0..255 float4 slots
            int row  = idx4 >> 2;                // 4 float4 per 16-float row
            int c4   = (idx4 & 3) * 4;
            const float* gp = A + (long long)(tileM + row) * N512 + k0 + c4;
            v2f lo = *(const v2f*)gp;
            v2f hi = *(const v2f*)(gp + 2);
            v2f* lp = (v2f*)&ldsA[row * LDA_PAD + c4];
            lp[0] = lo;
            lp[1] = hi;
        }
        #pragma unroll
        for (int rep = 0; rep < 2; ++rep) {
            int idx4 = tid + rep * 128;
            int row  = idx4 >> 4;                // 16 float4 per 64-float row
            int c4   = (idx4 & 15) * 4;
            const float* gp = B + (long long)(k0 + row) * N512 + tileN + c4;
            *(v4f*)&ldsB[row * 64 + c4] = *(const v4f*)gp;
        }
        if (k0 + 16 < N512) {
            __builtin_prefetch(A + (long long)(tileM + (tid & 63)) * N512 + k0 + 16, 0, 3);
            __builtin_prefetch(B + (long long)(k0 + 16 + (tid & 15)) * N512 + tileN, 0, 3);
        }
        __syncthreads();
#endif

        #pragma unroll
        for (int kk = 0; kk < 16; kk += 4) {
            // A fragments (ISA 7.12.2: lanes 0-15 -> K {kk,kk+1}; 16-31 -> {kk+2,kk+3})
            v2f a0 = *(const v2f*)&ldsA[(wm + lm)      * LDA_PAD + kk + 2 * half];
            v2f a1 = *(const v2f*)&ldsA[(wm + 16 + lm) * LDA_PAD + kk + 2 * half];
            // B fragments: VGPR0 holds K rows {kk | kk+2}, VGPR1 holds {kk+1 | kk+3}
            v2f b0, b1;
            b0.x = ldsB[(kk + 2 * half)     * 64 + wn + lm];
            b0.y = ldsB[(kk + 2 * half + 1) * 64 + wn + lm];
            b1.x = ldsB[(kk + 2 * half)     * 64 + wn + 16 + lm];
            b1.y = ldsB[(kk + 2 * half + 1) * 64 + wn + 16 + lm];

            acc00 = __builtin_amdgcn_wmma_f32_16x16x4_f32(false, a0, false, b0, (short)0, acc00, false, false);
            acc01 = __builtin_amdgcn_wmma_f32_16x16x4_f32(false, a0, false, b1, (short)0, acc01, false, false);
            acc10 = __builtin_amdgcn_wmma_f32_16x16x4_f32(false, a1, false, b0, (short)0, acc10, false, false);
            acc11 = __builtin_amdgcn_wmma_f32_16x16x4_f32(false, a1, false, b1, (short)0, acc11, false, false);
        }
        __syncthreads();
    }

    // ---- epilogue: C/D layout: VGPR r -> row r (lanes 0-15) / row r+8 (16-31)
    auto store_tile = [&](const v8f& av, int br, int bc) {
        #pragma unroll
        for (int r = 0; r < 8; ++r) {
            float v = alpha * av[r];
            long long off = (long long)(br + r + 8 * half) * N512 + bc + lm;
            if (accum) D[off] += v; else D[off] = v;
        }
    };
    store_tile(acc00, tileM + wm,      tileN + wn);
    store_tile(acc01, tileM + wm,      tileN + wn + 16);
    store_tile(acc10, tileM + wm + 16, tileN + wn);
    store_tile(acc11, tileM + wm + 16, tileN + wn + 16);
}

// ===========================================================================
// DFT tables:  C[u,k] = cos(2*pi*u*k/512),  S[u,k] = sin(2*pi*u*k/512)
// ===========================================================================
__global__ void dft_init_kernel(float* __restrict__ Ct, float* __restrict__ St)
{
    int idx = blockIdx.x * blockDim.x + threadIdx.x;     // 512*512
    int u = idx >> 9, k = idx & 511;
    int m = (u * k) & 511;
    float ang = (6.28318530717958647692f / 512.0f) * (float)m;
    Ct[idx] = cosf(ang);
    St[idx] = sinf(ang);
}

__global__ void zero_kernel(float* __restrict__ p, long long n)
{
    long long idx = (long long)blockIdx.x * blockDim.x + threadIdx.x;
    if (idx < n) p[idx] = 0.0f;
}

// ===========================================================================
// HyPaNet: mu[k] = softplus(W3*relu(W2*relu(W1*1+b1)+b2)+b3) + 1   (64 threads)
// ===========================================================================
__global__ void hypanet_kernel(const float* __restrict__ w1, const float* __restrict__ b1,
                               const float* __restrict__ w2, const float* __restrict__ b2,
                               const float* __restrict__ w3, const float* __restrict__ b3,
                               float* __restrict__ mu)
{
    __shared__ float h1[64], h2[64];
    int t = threadIdx.x;
    float v = w1[t] + b1[t];
    h1[t] = v > 0.0f ? v : 0.0f;
    __syncthreads();
    float s = b2[t];
    for (int j = 0; j < 64; ++j) s += w2[t * 64 + j] * h1[j];
    h2[t] = s > 0.0f ? s : 0.0f;
    __syncthreads();
    if (t < K1_ITERS) {
        float s3 = b3[t];
        for (int j = 0; j < 64; ++j) s3 += w3[t * 64 + j] * h2[j];
        float sp = (s3 > 20.0f) ? s3 : log1pf(expf(s3));
        mu[t] = sp + 1.0f;
    }
}

// ===========================================================================
// Fused 3x HyParNet: conv3x3(SAME,zero-pad) + BN(inference) + sigmoid + eps
// ===========================================================================
__global__ void hyparnet_kernel(const float* __restrict__ O,
    const float* __restrict__ w1, const float* __restrict__ c1b, const float* __restrict__ g1,
    const float* __restrict__ be1, const float* __restrict__ rm1, const float* __restrict__ rv1,
    const float* __restrict__ w2, const float* __restrict__ c2b, const float* __restrict__ g2,
    const float* __restrict__ be2, const float* __restrict__ rm2, const float* __restrict__ rv2,
    const float* __restrict__ w3, const float* __restrict__ c3b, const float* __restrict__ g3,
    const float* __restrict__ be3, const float* __restrict__ rm3, const float* __restrict__ rv3,
    float* __restrict__ alpha, float* __restrict__ px, float* __restrict__ py)
{
    long long idx = (long long)blockIdx.x * blockDim.x + threadIdx.x;  // < 8*512*512
    int b = (int)(idx >> 18);
    int rem = (int)(idx & 262143);
    int h = rem >> 9, w = rem & 511;
    const float* img = O + (long long)b * PLANE;

    float n[9];
    #pragma unroll
    for (int dy = 0; dy < 3; ++dy) {
        #pragma unroll
        for (int dx = 0; dx < 3; ++dx) {
            int hh = h + dy - 1, ww = w + dx - 1;
            float val = 0.0f;
            if (hh >= 0 && hh < 512 && ww >= 0 && ww < 512) val = img[hh * 512 + ww];
            n[dy * 3 + dx] = val;
        }
    }
    auto head = [&](const float* wk, const float* bk, const float* gk,
                    const float* bek, const float* rmk, const float* rvk) -> float {
        float y = bk[0];
        #pragma unroll
        for (int i = 0; i < 9; ++i) y += wk[i] * n[i];
        y = (y - rmk[0]) * rsqrtf(rvk[0] + BN_EPS_C);
        y = y * gk[0] + bek[0];
        return 1.0f / (1.0f + expf(-y)) + 0.001f;
    };
    alpha[idx] = 0.001f * head(w1, c1b, g1, be1, rm1, rv1);
    px[idx]    = head(w2, c2b, g2, be2, rm2, rv2);
    py[idx]    = head(w3, c3b, g3, be3, rm3, rv3);
}

// ===========================================================================
// rhs = O - mu_i * [ (d1+y1)[h-1,w] - (d1+y1)[h,w] + (d2+y2)[h,w-1] - (d2+y2)[h,w] ]
// ===========================================================================
__global__ void rhs_kernel(const float* __restrict__ O,
                           const float* __restrict__ d1, const float* __restrict__ d2,
                           const float* __restrict__ y1, const float* __restrict__ y2,
                           const float* __restrict__ mu, int it, float* __restrict__ rhs)
{
    long long idx = (long long)blockIdx.x * blockDim.x + threadIdx.x;
    int b = (int)(idx >> 18);
    int rem = (int)(idx & 262143);
    int h = rem >> 9, w = rem & 511;
    float mui = mu[it];
    long long base = (long long)b * PLANE;
    int hm = (h == 0) ? 511 : h - 1;
    int wm = (w == 0) ? 511 : w - 1;
    long long p  = base + (long long)h * 512 + w;
    long long pu = base + (long long)hm * 512 + w;
    long long pl = base + (long long)h * 512 + wm;
    float u  = d1[p]  + y1[p];
    float uu = d1[pu] + y1[pu];
    float v  = d2[p]  + y2[p];
    float vl = d2[pl] + y2[pl];
    rhs[idx] = O[idx] - mui * ((uu - u) + (vl - v));
}

// ===========================================================================
// Spectral scale: divide (Yre, Yim') by 1 + mu*(4 sin^2(pi u/N) + 4 sin^2(pi v/N))
// ===========================================================================
__global__ void scale_kernel(float* __restrict__ Tre, float* __restrict__ Tim,
                             const float* __restrict__ mu, int it)
{
    long long idx = (long long)blockIdx.x * blockDim.x + threadIdx.x;
    int rem = (int)(idx & 262143);
    int u = rem >> 9, v = rem & 511;
    float mui = mu[it];
    const float c = 3.14159265358979323846f / 512.0f;
    float su = __sinf(c * (float)u);
    float sv = __sinf(c * (float)v);
    float g = 1.0f / (1.0f + mui * 4.0f * (su * su + sv * sv));
    Tre[idx] *= g;
    Tim[idx] *= g;
}

// ===========================================================================
// Shrinkage + dual update (in place on d1,d2,y1,y2)
// ===========================================================================
__global__ void update_kernel(const float* __restrict__ I,
                              const float* __restrict__ alpha,
                              const float* __restrict__ px, const float* __restrict__ py,
                              float* __restrict__ d1, float* __restrict__ d2,
                              float* __restrict__ y1, float* __restrict__ y2,
                              const float* __restrict__ mu, int it)
{
    long long idx = (long long)blockIdx.x * blockDim.x + threadIdx.x;
    int b = (int)(idx >> 18);
    int rem = (int)(idx & 262143);
    int h = rem >> 9, w = rem & 511;
    float mui = mu[it];
    long long base = (long long)b * PLANE;
    int hp = (h == 511) ? 0 : h + 1;
    int wp = (w == 511) ? 0 : w + 1;
    float Ic = I[idx];
    float Dx = I[base + (long long)hp * 512 + w] - Ic;
    float Dy = I[base + (long long)h * 512 + wp] - Ic;
    float a  = alpha[idx];
    float ex = 2.0f - px[idx];
    float ey = 2.0f - py[idx];
    float wx = 1.0f / fmaxf(powf(fabsf(Dx), ex), EPS_C);
    float wy = 1.0f / fmaxf(powf(fabsf(Dy), ey), EPS_C);
    float y1v = y1[idx], y2v = y2[idx];
    float d1n = mui * (Dx - y1v) / (2.0f * a * wx + mui);
    float d2n = mui * (Dy - y2v) / (2.0f * a * wy + mui);
    d1[idx] = d1n;
    d2[idx] = d2n;
    y1[idx] = y1v + (d1n - Dx);
    y2[idx] = y2v + (d2n - Dy);
}

__global__ void final_r_kernel(const float* __restrict__ O, const float* __restrict__ Ilast,
                               float* __restrict__ R)
{
    long long idx = (long long)blockIdx.x * blockDim.x + threadIdx.x;
    R[idx] = O[idx] / (Ilast[idx] + EPS_C);
}

// ===========================================================================
// Host orchestration
// ===========================================================================
extern "C" void kernel_launch(void* const* d_in, const int* in_sizes, int n_in,
                              void* d_out, int out_size, void* d_ws, size_t ws_size,
                              hipStream_t stream)
{
    const float* O     = (const float*)d_in[0];
    const float* hp_w1 = (const float*)d_in[1];
    const float* hp_b1 = (const float*)d_in[2];
    const float* hp_w2 = (const float*)d_in[3];
    const float* hp_b2 = (const float*)d_in[4];
    const float* hp_w3 = (const float*)d_in[5];
    const float* hp_b3 = (const float*)d_in[6];
    const float* p1[6] = { (const float*)d_in[7],  (const float*)d_in[8],  (const float*)d_in[9],
                           (const float*)d_in[10], (const float*)d_in[11], (const float*)d_in[12] };
    const float* p2[6] = { (const float*)d_in[13], (const float*)d_in[14], (const float*)d_in[15],
                           (const float*)d_in[16], (const float*)d_in[17], (const float*)d_in[18] };
    const float* p3[6] = { (const float*)d_in[19], (const float*)d_in[20], (const float*)d_in[21],
                           (const float*)d_in[22], (const float*)d_in[23], (const float*)d_in[24] };

    float* out    = (float*)d_out;
    float* Istack = out;                       // 10 * IMGSET
    float* R      = out + 10 * IMGSET;
    float* alpha  = out + 11 * IMGSET;
    float* px     = out + 12 * IMGSET;
    float* py     = out + 13 * IMGSET;
    float* mu     = out + 14 * IMGSET;         // 10 floats

    float* ws   = (float*)d_ws;
    float* dftC = ws;
    float* dftS = dftC + PLANE;
    float* d1   = dftS + PLANE;
    float* d2   = d1 + IMGSET;
    float* y1   = d2 + IMGSET;
    float* y2   = y1 + IMGSET;
    float* RHS  = y2 + IMGSET;
    float* T1   = RHS + IMGSET;
    float* T2   = T1 + IMGSET;
    float* T3   = T2 + IMGSET;
    float* T4   = T3 + IMGSET;

    const int PTB = 256;
    const int planeBlocks = (int)(PLANE / PTB);          // 1024
    const int imgBlocks   = (int)(IMGSET / PTB);         // 8192
    const long long zn    = 4 * IMGSET;
    const int zeroBlocks  = (int)((zn + PTB - 1) / PTB);

    dft_init_kernel<<<planeBlocks, PTB, 0, stream>>>(dftC, dftS);
    zero_kernel<<<zeroBlocks, PTB, 0, stream>>>(d1, zn);   // zeros d1,d2,y1,y2 (contiguous)
    hypanet_kernel<<<1, 64, 0, stream>>>(hp_w1, hp_b1, hp_w2, hp_b2, hp_w3, hp_b3, mu);
    hyparnet_kernel<<<imgBlocks, PTB, 0, stream>>>(O,
        p1[0], p1[1], p1[2], p1[3], p1[4], p1[5],
        p2[0], p2[1], p2[2], p2[3], p2[4], p2[5],
        p3[0], p3[1], p3[2], p3[3], p3[4], p3[5],
        alpha, px, py);

    dim3 gg(8, 8, 8);            // 512/64 x 512/64 x batch
    const float invN2 = 1.0f / 262144.0f;

    for (int it = 0; it < K1_ITERS; ++it) {
        rhs_kernel<<<imgBlocks, PTB, 0, stream>>>(O, d1, d2, y1, y2, mu, it, RHS);

        // Forward DFT:  x*F = A - iB  (A = x*C, B = x*S ; C,S symmetric)
        gemm512_wmma<<<gg, 128, 0, stream>>>(T1, RHS, dftC, 1.0f, 0, PLANE, PLANE, 0);
        gemm512_wmma<<<gg, 128, 0, stream>>>(T2, RHS, dftS, 1.0f, 0, PLANE, PLANE, 0);
        // Yre = C*A - S*B ;  Yim' (= -Yim) = C*B + S*A
        gemm512_wmma<<<gg, 128, 0, stream>>>(T3, dftC, T1,  1.0f, 0, PLANE, 0, PLANE);
        gemm512_wmma<<<gg, 128, 0, stream>>>(T3, dftS, T2, -1.0f, 1, PLANE, 0, PLANE);
        gemm512_wmma<<<gg, 128, 0, stream>>>(T4, dftC, T2,  1.0f, 0, PLANE, 0, PLANE);
        gemm512_wmma<<<gg, 128, 0, stream>>>(T4, dftS, T1,  1.0f, 1, PLANE, 0, PLANE);
        // divide by (1 + mu * eigDtD)
        scale_kernel<<<imgBlocks, PTB, 0, stream>>>(T3, T4, mu, it);
        // Inverse: P = C*Zre + S*Zim' ;  Qn = C*Zim' - S*Zre ;  I = (P*C + Qn*S)/N^2
        gemm512_wmma<<<gg, 128, 0, stream>>>(T1, dftC, T3,  1.0f, 0, PLANE, 0, PLANE);
        gemm512_wmma<<<gg, 128, 0, stream>>>(T1, dftS, T4,  1.0f, 1, PLANE, 0, PLANE);
        gemm512_wmma<<<gg, 128, 0, stream>>>(T2, dftC, T4,  1.0f, 0, PLANE, 0, PLANE);
        gemm512_wmma<<<gg, 128, 0, stream>>>(T2, dftS, T3, -1.0f, 1, PLANE, 0, PLANE);

        float* Iout = Istack + (long long)it * IMGSET;
        gemm512_wmma<<<gg, 128, 0, stream>>>(Iout, T1, dftC, invN2, 0, PLANE, PLANE, 0);
        gemm512_wmma<<<gg, 128, 0, stream>>>(Iout, T2, dftS, invN2, 1, PLANE, PLANE, 0);

        update_kernel<<<imgBlocks, PTB, 0, stream>>>(Iout, alpha, px, py,
                                                     d1, d2, y1, y2, mu, it);
    }

    final_r_kernel<<<imgBlocks, PTB, 0, stream>>>(O, Istack + 9 * IMGSET, R);
}